// AFH_12412455485723
// MI455X (gfx1250) — compile-verified
//
#include <hip/hip_runtime.h>
#include <hip/hip_bf16.h>

typedef float v2f __attribute__((ext_vector_type(2)));
typedef float v8f __attribute__((ext_vector_type(8)));

#define BATCH 64
#define CIN   1024
#define HW    196
#define MOUT  1024      // q(512) + k(512) output channels of W_qkv (V unused)
#define NT    13        // ceil(196/16)
#define MG    16        // M groups of 64 rows (4 WMMA tiles each)

// workspace layout (in floats)
#define QK_OFF    0
#define QK_SZ     (BATCH * MOUT * HW)     // 12,845,056 floats (51.4 MB)
#define QSUM_OFF  (QK_OFF + QK_SZ)
#define QSUM_SZ   (BATCH * 512)
#define AW_OFF    (QSUM_OFF + QSUM_SZ)
#define AW_SZ     (BATCH * 64 * 14)
#define AH_OFF    (AW_OFF + AW_SZ)
#define AH_SZ     (BATCH * 64 * 14)
#define ATTN_OFF  (AH_OFF + AH_SZ)
#define ATTN_SZ   (BATCH * HW)

// ---------------------------------------------------------------------------
// Kernel 1: Q/K projection GEMM with f32 WMMA (16x16x4).
// One wave computes a 64(M) x 16(N) macro-tile (4 accumulators), so the two
// B loads per K-step feed 4 WMMAs: 6 vmem / 4 wmma instead of 12 / 4.
// A (16x4 from W): lane<16 -> {K=k0,k0+1}, lane>=16 -> {K=k0+2,k0+3}, row M=lane&15.
// B (4x16 from x): symmetric, col N=lane&15.
// D: lane l holds N=l&15, VGPR r holds M = r + 8*(lane>>4).
// ---------------------------------------------------------------------------
__global__ __launch_bounds__(256)
void qk_gemm_wmma(const float* __restrict__ x, const float* __restrict__ Wq,
                  const float* __restrict__ bias, float* __restrict__ qk)
{
    const int wid  = blockIdx.x * 8 + (threadIdx.x >> 5);
    const int lane = threadIdx.x & 31;
    const int nt   = wid % NT;
    const int tmp  = wid / NT;
    const int mg   = tmp % MG;                 // 64-row M group
    const int b    = tmp / MG;
    const int half = lane >> 4;
    const int l    = lane & 15;

    const int mbase = mg * 64 + l;             // W_qkv row for A operand, tile j adds j*16
    int n = nt * 16 + l;                       // spatial position
    if (n > HW - 1) n = HW - 1;                // clamp loads; stores are guarded

    const float* __restrict__ A0 = Wq + (size_t)(mbase)      * CIN;
    const float* __restrict__ A1 = Wq + (size_t)(mbase + 16) * CIN;
    const float* __restrict__ A2 = Wq + (size_t)(mbase + 32) * CIN;
    const float* __restrict__ A3 = Wq + (size_t)(mbase + 48) * CIN;
    const float* __restrict__ Bbase = x + (size_t)b * CIN * HW + n;

    v8f acc0 = {}, acc1 = {}, acc2 = {}, acc3 = {};
    #pragma unroll 2
    for (int k0 = 0; k0 < CIN; k0 += 4) {
        const int kb = k0 + 2 * half;
        v2f bb;
        bb.x = Bbase[(size_t)kb * HW];
        bb.y = Bbase[(size_t)(kb + 1) * HW];
        v2f a0 = *(const v2f*)(A0 + kb);       // 8B aligned (kb even)
        v2f a1 = *(const v2f*)(A1 + kb);
        v2f a2 = *(const v2f*)(A2 + kb);
        v2f a3 = *(const v2f*)(A3 + kb);
        acc0 = __builtin_amdgcn_wmma_f32_16x16x4_f32(false, a0, false, bb, (short)0, acc0, false, false);
        acc1 = __builtin_amdgcn_wmma_f32_16x16x4_f32(false, a1, false, bb, (short)0, acc1, false, false);
        acc2 = __builtin_amdgcn_wmma_f32_16x16x4_f32(false, a2, false, bb, (short)0, acc2, false, false);
        acc3 = __builtin_amdgcn_wmma_f32_16x16x4_f32(false, a3, false, bb, (short)0, acc3, false, false);
    }

    const int p = nt * 16 + l;
    if (p < HW) {
        float* __restrict__ outb = qk + (size_t)b * MOUT * HW + p;
        #pragma unroll
        for (int j = 0; j < 4; ++j) {
            const v8f acc = (j == 0) ? acc0 : (j == 1) ? acc1 : (j == 2) ? acc2 : acc3;
            #pragma unroll
            for (int r = 0; r < 8; ++r) {
                const int o = mg * 64 + j * 16 + r + 8 * half;
                float v = acc[r] + bias[o];
                if (o < 512) v *= 0.125f;      // q scale: dkh^-0.5 = 1/8
                outb[(size_t)o * HW] = v;
            }
        }
    }
}

// ---------------------------------------------------------------------------
// Kernel 2: per-(b,head,pos) normalize q over d (64), then sum qhat over the
// 196 positions -> qsum[b, n, d].  One block per (b, head).
// ---------------------------------------------------------------------------
__global__ __launch_bounds__(256)
void norm_qsum(const float* __restrict__ qk, float* __restrict__ qsum)
{
    __shared__ float rn[HW];
    const int b = blockIdx.x >> 3;
    const int n = blockIdx.x & 7;
    const float* __restrict__ q = qk + ((size_t)b * MOUT + n * 64) * HW;
    const int tid = threadIdx.x;
    if (tid < HW) {
        float ss = 0.f;
        #pragma unroll 8
        for (int d = 0; d < 64; ++d) { float v = q[d * HW + tid]; ss += v * v; }
        rn[tid] = rsqrtf(ss);
    }
    __syncthreads();
    if (tid < 64) {
        float s = 0.f;
        for (int p = 0; p < HW; ++p) s += q[tid * HW + p] * rn[p];
        qsum[blockIdx.x * 64 + tid] = s;
    }
}

// ---------------------------------------------------------------------------
// Kernel 3: row/column sums of raw q (over heads too) for the relative-logit
// means.  aw[b,d,wy] = sum_{n,hx} q[b,n,d,hx,wy]; ah[b,d,hx] = sum_{n,wy} ...
// ---------------------------------------------------------------------------
__global__ __launch_bounds__(256)
void rel_colrow_sums(const float* __restrict__ qk, float* __restrict__ aw,
                     float* __restrict__ ah)
{
    const int b = blockIdx.x;
    const float* __restrict__ q = qk + (size_t)b * MOUT * HW;   // q channels 0..511
    for (int idx = threadIdx.x; idx < 64 * 14; idx += blockDim.x) {
        const int d = idx / 14;
        const int t = idx % 14;
        float sw = 0.f, sh = 0.f;
        for (int n = 0; n < 8; ++n) {
            const float* __restrict__ row = q + (size_t)(n * 64 + d) * HW;
            #pragma unroll
            for (int u = 0; u < 14; ++u) {
                sw += row[u * 14 + t];     // column wy=t, sum over hx
                sh += row[t * 14 + u];     // row    hx=t, sum over wy
            }
        }
        aw[(b * 64 + d) * 14 + t] = sw;
        ah[(b * 64 + d) * 14 + t] = sh;
    }
}

// ---------------------------------------------------------------------------
// Kernel 4: mean logits + softmax.  mean_logits[b,kh,kw] =
//   (1/8) sum_{n,d} qbar[n,d]*khat[n,d,kp]  + mh[b,kh] + mw[b,kw]
// with qbar = 0.5*(qsum[pair[b]]+qsum[b])/196, rel_to_abs => m = k - y + 13.
// One block (256 threads) per batch.
// ---------------------------------------------------------------------------
__global__ __launch_bounds__(256)
void logits_softmax(const float* __restrict__ qk, const float* __restrict__ qsum,
                    const float* __restrict__ aw, const float* __restrict__ ah,
                    const float* __restrict__ relw, const float* __restrict__ relh,
                    const int* __restrict__ pair, float* __restrict__ attn)
{
    __shared__ float qbar[512];
    __shared__ float mwv[16], mhv[16];
    __shared__ float red[256];
    const int b   = blockIdx.x;
    const int tid = threadIdx.x;
    const int pb  = pair[b];

    for (int i = tid; i < 512; i += 256)
        qbar[i] = (qsum[pb * 512 + i] + qsum[b * 512 + i]) * (0.5f / (196.f * 8.f));

    if (tid < 14) {                                  // mw[b, kw=tid]
        const int kw = tid;
        float s = 0.f;
        const float* __restrict__ a = aw + b * 64 * 14;
        for (int wy = 0; wy < 14; ++wy) {
            const float* __restrict__ r = relw + (kw - wy + 13) * 64;
            for (int d = 0; d < 64; ++d) s += a[d * 14 + wy] * r[d];
        }
        mwv[kw] = s * (1.f / 1568.f);                // 1/(8*196)
    } else if (tid >= 32 && tid < 46) {              // mh[b, kh=tid-32]
        const int kh = tid - 32;
        float s = 0.f;
        const float* __restrict__ a = ah + b * 64 * 14;
        for (int hx = 0; hx < 14; ++hx) {
            const float* __restrict__ r = relh + (kh - hx + 13) * 64;
            for (int d = 0; d < 64; ++d) s += a[d * 14 + hx] * r[d];
        }
        mhv[kh] = s * (1.f / 1568.f);
    }
    __syncthreads();

    float ml = -3.402823466e38f;
    if (tid < HW) {
        const float* __restrict__ kb = qk + ((size_t)b * MOUT + 512) * HW + tid;
        float dot = 0.f;
        for (int n = 0; n < 8; ++n) {
            float ss = 0.f, dt = 0.f;
            #pragma unroll 8
            for (int d = 0; d < 64; ++d) {
                const float kv = kb[(size_t)(n * 64 + d) * HW];
                ss += kv * kv;
                dt += qbar[n * 64 + d] * kv;
            }
            dot += dt * rsqrtf(ss);                  // normalize k on the fly
        }
        ml = dot + mhv[tid / 14] + mwv[tid % 14];
    }

    red[tid] = ml;
    __syncthreads();
    for (int s = 128; s > 0; s >>= 1) {
        if (tid < s) red[tid] = fmaxf(red[tid], red[tid + s]);
        __syncthreads();
    }
    const float mx = red[0];
    __syncthreads();
    const float e = (tid < HW) ? expf(ml - mx) : 0.f;
    red[tid] = e;
    __syncthreads();
    for (int s = 128; s > 0; s >>= 1) {
        if (tid < s) red[tid] += red[tid + s];
        __syncthreads();
    }
    const float inv = 1.f / red[0];
    if (tid < HW) attn[b * HW + tid] = e * inv;
}

// ---------------------------------------------------------------------------
// Kernel 5: streaming epilogue — out = concat(x/49, x*attn), float4 I/O.
// ---------------------------------------------------------------------------
__global__ __launch_bounds__(256)
void write_out(const float* __restrict__ x, const float* __restrict__ attn,
               float* __restrict__ out)
{
    const size_t total4 = (size_t)BATCH * 2048 * HW / 4;
    const size_t i = (size_t)blockIdx.x * blockDim.x + threadIdx.x;
    if (i >= total4) return;
    const size_t flat = i * 4;
    const int b   = (int)(flat / ((size_t)2048 * HW));
    const int rem = (int)(flat % ((size_t)2048 * HW));
    const int ch  = rem / HW;
    const int p   = rem % HW;                        // multiple of 4 (196 % 4 == 0)
    float4 v;
    if (ch < 1024) {
        v = *(const float4*)(x + ((size_t)b * 1024 + ch) * HW + p);
        const float s = 1.f / 49.f;
        v.x *= s; v.y *= s; v.z *= s; v.w *= s;
    } else {
        v = *(const float4*)(x + ((size_t)b * 1024 + (ch - 1024)) * HW + p);
        const float4 a = *(const float4*)(attn + b * HW + p);
        v.x *= a.x; v.y *= a.y; v.z *= a.z; v.w *= a.w;
    }
    *(float4*)(out + flat) = v;
}

// ---------------------------------------------------------------------------
extern "C" void kernel_launch(void* const* d_in, const int* in_sizes, int n_in,
                              void* d_out, int out_size, void* d_ws, size_t ws_size,
                              hipStream_t stream)
{
    const float* x     = (const float*)d_in[0];   // [64,1024,14,14]
    const float* W_qkv = (const float*)d_in[1];   // [1536,1024]
    const float* b_qkv = (const float*)d_in[2];   // [1536]
    const float* relw  = (const float*)d_in[3];   // [27,64]
    const float* relh  = (const float*)d_in[4];   // [27,64]
    const int*   pair  = (const int*)d_in[5];     // [64]
    float* out = (float*)d_out;

    float* ws   = (float*)d_ws;
    float* qk   = ws + QK_OFF;
    float* qsum = ws + QSUM_OFF;
    float* aw   = ws + AW_OFF;
    float* ah   = ws + AH_OFF;
    float* attn = ws + ATTN_OFF;

    // 1) WMMA GEMM: 13312 waves = 1664 blocks of 8 waves, 64x16 tile per wave
    qk_gemm_wmma<<<(BATCH * MG * NT) / 8, 256, 0, stream>>>(x, W_qkv, b_qkv, qk);
    // 2) q normalization + position sums
    norm_qsum<<<BATCH * 8, 256, 0, stream>>>(qk, qsum);
    // 3) relative-logit row/col sums
    rel_colrow_sums<<<BATCH, 256, 0, stream>>>(qk, aw, ah);
    // 4) mean logits + softmax
    logits_softmax<<<BATCH, 256, 0, stream>>>(qk, qsum, aw, ah, relw, relh, pair, attn);
    // 5) epilogue
    const size_t total4 = (size_t)BATCH * 2048 * HW / 4;
    write_out<<<(unsigned)((total4 + 255) / 256), 256, 0, stream>>>(x, attn, out);
}